// S4BlockInterface_63367947485689
// MI455X (gfx1250) — compile-verified
//
#include <hip/hip_runtime.h>

// ---- problem sizes (match reference) ----
#define DMODEL 512
#define NHID   64
#define BATCH  8
#define LSEQ   4096
#define TCH    16                 // chunk length (WMMA tile M)
#define NCHUNK (LSEQ / TCH)       // 256
#define WPB    8                  // waves (=channels) per block
#define NBLK   (DMODEL / WPB)     // 64 blocks
#define NTHR   (WPB * 32)         // 256 threads, wave32

typedef float v2f __attribute__((ext_vector_type(2)));
typedef float v8f __attribute__((ext_vector_type(8)));

// LDS layout (floats). Two phases share the same 40KB region:
//   main phase:  sX [16t][8b][8d]=1024 | sY 1024 | sH [8w][64n][16c]=8192
//   init phase:  sPow [8w][17t][64n]=8704 | sC 512 | sBB 512 | sK 128
#define OFF_X   0
#define OFF_Y   1024
#define OFF_H   2048
#define SMEM_F  10240
#define OFF_POW 0
#define OFF_C   8704
#define OFF_BB  9216
#define OFF_K   9728

static __device__ __forceinline__ v8f wmma_k4(v2f a, v2f b, v8f c) {
  // D(16x16,f32) = A(16x4,f32) * B(4x16,f32) + C ; 8-arg VOP3P form
  return __builtin_amdgcn_wmma_f32_16x16x4_f32(false, a, false, b,
                                               (short)0, c, false, false);
}

__global__ __launch_bounds__(NTHR)
void s4d_chunked_wmma(const float* __restrict__ x,
                      const float* __restrict__ A,
                      const float* __restrict__ Bm,
                      const float* __restrict__ C,
                      const float* __restrict__ D,
                      const float* __restrict__ dt,
                      float* __restrict__ out)
{
  __shared__ float smem[SMEM_F];

  const int tid  = threadIdx.x;
  const int w    = tid >> 5;        // wave id = channel within block
  const int lane = tid & 31;
  const int half = lane >> 4;       // A-operand K half (ISA 16x4 f32 layout)
  const int li   = lane & 15;       // row (A) / column (B,C,D) index
  const int d    = blockIdx.x * WPB + w;
  const int d0   = blockIdx.x * WPB;
  const float dtv = dt[0];
  const float Dd  = D[d];

  // ================= init phase: discretize + power tables ================
  float* sPow = smem + OFF_POW + w * (TCH + 1) * NHID;   // a^t, t=0..16
  float* sCc  = smem + OFF_C   + w * NHID;
  float* sBB  = smem + OFF_BB  + w * NHID;
  float* sK   = smem + OFF_K   + w * TCH;

  #pragma unroll
  for (int j = 0; j < 2; ++j) {
    const int n = lane + 32 * j;
    const float Av = A[d * NHID + n];
    const float a  = expf(dtv * Av);              // ZOH: A_bar
    sCc[n] = C[d * NHID + n];
    sBB[n] = (a - 1.0f) / Av * Bm[d * NHID + n];  // B_bar
    float p = 1.0f;
    for (int t = 0; t <= TCH; ++t) { sPow[t * NHID + n] = p; p *= a; }
  }
  __syncthreads();

  // Toeplitz conv kernel K[tau] = sum_n C*bb*a^tau
  {
    float acc = 0.f;
    for (int n = 0; n < NHID; ++n)
      acc += sCc[n] * sBB[n] * sPow[li * NHID + n];
    if (half == 0) sK[li] = acc;
  }
  __syncthreads();

  // ------- constant A-operand fragments (resident in VGPRs) -------
  v2f m1f[4];                       // Toeplitz [16t x 16s], lower-tri
  #pragma unroll
  for (int kc = 0; kc < 4; ++kc) {
    const int s0 = 4 * kc + 2 * half;
    m1f[kc].x = (li >= s0    ) ? sK[li - s0    ] : 0.f;
    m1f[kc].y = (li >= s0 + 1) ? sK[li - s0 - 1] : 0.f;
  }
  v2f m2f[16];                      // state->y  [16t x 64n] : C_n * a^(t+1)
  #pragma unroll
  for (int kc = 0; kc < 16; ++kc) {
    const int n0 = 4 * kc + 2 * half;
    m2f[kc].x = sCc[n0    ] * sPow[(li + 1) * NHID + n0    ];
    m2f[kc].y = sCc[n0 + 1] * sPow[(li + 1) * NHID + n0 + 1];
  }
  v2f m3f[4][4];                    // x->state  [64n x 16s] : bb_n * a^(15-s)
  #pragma unroll
  for (int rt = 0; rt < 4; ++rt) {
    const int n = 16 * rt + li;
    #pragma unroll
    for (int kc = 0; kc < 4; ++kc) {
      const int s0 = 4 * kc + 2 * half;
      m3f[rt][kc].x = sBB[n] * sPow[(15 - s0) * NHID + n];
      m3f[rt][kc].y = sBB[n] * sPow[(14 - s0) * NHID + n];
    }
  }
  float aT[4][8];                   // a^16 at this lane's C-tile rows
  #pragma unroll
  for (int rt = 0; rt < 4; ++rt)
    #pragma unroll
    for (int r = 0; r < 8; ++r)
      aT[rt][r] = sPow[TCH * NHID + (16 * rt + r + 8 * half)];
  __syncthreads();   // init region reads done; buffers re-used below

  // ================= main loop over 256 chunks ================
  float* sX = smem + OFF_X;                 // [t][b][dg]
  float* sY = smem + OFF_Y;
  float* sH = smem + OFF_H + w * NHID * 16; // [n][col] B-operand staging

  v8f ht[4];                                // state h as 4 C-tiles (64n x 8b)
  #pragma unroll
  for (int rt = 0; rt < 4; ++rt)
    #pragma unroll
    for (int r = 0; r < 8; ++r) ht[rt][r] = 0.f;

  const int bcol = li & 7;

  for (int c = 0; c < NCHUNK; ++c) {
    // ---- stage x chunk (d-contiguous 32B runs) ----
    #pragma unroll
    for (int i = 0; i < (TCH * BATCH * WPB) / NTHR; ++i) {
      const int e  = tid + i * NTHR;
      const int t  = e >> 6;
      const int b  = (e >> 3) & 7;
      const int dg = e & 7;
      sX[e] = x[((size_t)b * LSEQ + (size_t)c * TCH + t) * DMODEL + d0 + dg];
    }
    // ---- publish h0 tiles for B-operand reads ----
    #pragma unroll
    for (int rt = 0; rt < 4; ++rt)
      #pragma unroll
      for (int r = 0; r < 8; ++r)
        sH[(16 * rt + r + 8 * half) * 16 + li] = ht[rt][r];
    __syncthreads();

    // ---- x chunk as B-operand fragments [16s x 16cols(b)] ----
    v2f xb[4];
    #pragma unroll
    for (int kc = 0; kc < 4; ++kc) {
      const int s0 = 4 * kc + 2 * half;
      xb[kc].x = sX[((s0    ) * BATCH + bcol) * WPB + w];
      xb[kc].y = sX[((s0 + 1) * BATCH + bcol) * WPB + w];
    }

    // ---- y tile = Toeplitz conv + state propagation (one accumulator) ----
    v8f yt;
    #pragma unroll
    for (int r = 0; r < 8; ++r) yt[r] = 0.f;
    #pragma unroll
    for (int kc = 0; kc < 4; ++kc) yt = wmma_k4(m1f[kc], xb[kc], yt);
    #pragma unroll
    for (int kc = 0; kc < 16; ++kc) {
      const int n0 = 4 * kc + 2 * half;
      v2f hb;
      hb.x = sH[(n0    ) * 16 + li];
      hb.y = sH[(n0 + 1) * 16 + li];
      yt = wmma_k4(m2f[kc], hb, yt);
    }

    // ---- state update: h_new = a^16 ∘ h0 + M3 · Xc ----
    #pragma unroll
    for (int rt = 0; rt < 4; ++rt) {
      v8f acc;
      #pragma unroll
      for (int r = 0; r < 8; ++r) acc[r] = aT[rt][r] * ht[rt][r];
      #pragma unroll
      for (int kc = 0; kc < 4; ++kc) acc = wmma_k4(m3f[rt][kc], xb[kc], acc);
      ht[rt] = acc;
    }

    // ---- y (+ D*x skip) to LDS; only cols 0..7 carry valid batch ----
    if (li < BATCH) {
      #pragma unroll
      for (int r = 0; r < 8; ++r) {
        const int t = r + 8 * half;
        const int e = (t * BATCH + li) * WPB + w;
        sY[e] = yt[r] + Dd * sX[e];
      }
    }
    // prefetch next chunk's x lines into cache
    if (c + 1 < NCHUNK)
      __builtin_prefetch(&x[((size_t)(tid & 7) * LSEQ +
                             (size_t)(c + 1) * TCH + (tid >> 5)) * DMODEL + d0],
                         0, 1);
    __syncthreads();

    // ---- coalesced store of this chunk ----
    #pragma unroll
    for (int i = 0; i < (TCH * BATCH * WPB) / NTHR; ++i) {
      const int e  = tid + i * NTHR;
      const int t  = e >> 6;
      const int b  = (e >> 3) & 7;
      const int dg = e & 7;
      out[((size_t)b * LSEQ + (size_t)c * TCH + t) * DMODEL + d0 + dg] = sY[e];
    }
    // next iteration's writes (sX/sH) only conflict with reads already
    // fenced by the barrier above; no extra barrier needed here.
  }
}

extern "C" void kernel_launch(void* const* d_in, const int* in_sizes, int n_in,
                              void* d_out, int out_size, void* d_ws, size_t ws_size,
                              hipStream_t stream) {
  (void)in_sizes; (void)n_in; (void)d_ws; (void)ws_size; (void)out_size;
  const float* x  = (const float*)d_in[0];
  const float* A  = (const float*)d_in[1];
  const float* Bm = (const float*)d_in[2];
  const float* C  = (const float*)d_in[3];
  const float* D  = (const float*)d_in[4];
  const float* dt = (const float*)d_in[5];
  float* out = (float*)d_out;

  s4d_chunked_wmma<<<dim3(NBLK), dim3(NTHR), 0, stream>>>(x, A, Bm, C, D, dt, out);
}